// Mono_Flat_GLM2_49684181680769
// MI455X (gfx1250) — compile-verified
//
#include <hip/hip_runtime.h>
#include <hip/hip_bf16.h>
#include <math.h>

typedef __attribute__((ext_vector_type(2))) float v2f;
typedef __attribute__((ext_vector_type(4))) float v4f;
typedef __attribute__((ext_vector_type(8))) float v8f;

// ---------------- problem constants ----------------
#define SUB_NO  20
#define E_NO    2000
#define I_NO    500
#define T_NO    200
#define T_DATA  20000

#define NPAD    32        // subunit dim padded to 2 WMMA n-tiles
#define KPAD_E  2048      // E_NO padded to multiple of 64
#define KPAD_I  512       // I_NO padded to multiple of 64

// ---------------- setup: softmax B-matrices, squared kernels, exp weights, out_filters ----
__global__ __launch_bounds__(256) void setup_kernel(
    const float* __restrict__ W_e, const float* __restrict__ W_i,
    const float* __restrict__ Ce,  const float* __restrict__ Ci,
    const float* __restrict__ W_sub, const float* __restrict__ temp,
    float* __restrict__ Bbuf_e, float* __restrict__ Bbuf_i,
    float* __restrict__ We2, float* __restrict__ Wi2n,
    float* __restrict__ expW, float* __restrict__ out_filters)
{
    const int g = blockIdx.x * blockDim.x + threadIdx.x;
    const float invT = 1.0f / temp[0];

    if (g < KPAD_E) {
        const int e = g;
        if (e < E_NO) {
            float vals[SUB_NO];
            float m = -3.4e38f;
            #pragma unroll
            for (int s = 0; s < SUB_NO; ++s) {
                float v = Ce[s * E_NO + e] * invT;
                vals[s] = v; m = fmaxf(m, v);
            }
            float sum = 0.f;
            #pragma unroll
            for (int s = 0; s < SUB_NO; ++s) { float v = expf(vals[s] - m); vals[s] = v; sum += v; }
            const float inv = 1.0f / sum;
            #pragma unroll
            for (int s = 0; s < SUB_NO; ++s) Bbuf_e[e * NPAD + s] = vals[s] * inv;
            #pragma unroll
            for (int s = SUB_NO; s < NPAD; ++s) Bbuf_e[e * NPAD + s] = 0.f;
        } else {
            #pragma unroll
            for (int s = 0; s < NPAD; ++s) Bbuf_e[e * NPAD + s] = 0.f;
        }
    } else if (g < KPAD_E + KPAD_I) {
        const int e = g - KPAD_E;
        if (e < I_NO) {
            float vals[SUB_NO];
            float m = -3.4e38f;
            #pragma unroll
            for (int s = 0; s < SUB_NO; ++s) {
                float v = Ci[s * I_NO + e] * invT;
                vals[s] = v; m = fmaxf(m, v);
            }
            float sum = 0.f;
            #pragma unroll
            for (int s = 0; s < SUB_NO; ++s) { float v = expf(vals[s] - m); vals[s] = v; sum += v; }
            const float inv = 1.0f / sum;
            #pragma unroll
            for (int s = 0; s < SUB_NO; ++s) Bbuf_i[e * NPAD + s] = vals[s] * inv;
            #pragma unroll
            for (int s = SUB_NO; s < NPAD; ++s) Bbuf_i[e * NPAD + s] = 0.f;
        } else {
            #pragma unroll
            for (int s = 0; s < NPAD; ++s) Bbuf_i[e * NPAD + s] = 0.f;
        }
    } else if (g < KPAD_E + KPAD_I + SUB_NO * T_NO) {
        const int idx = g - (KPAD_E + KPAD_I);
        const float we = W_e[idx];
        const float v = we * we;
        We2[idx] = v;
        out_filters[idx] = v;                       // rows 0..19 of out_filters
        const float wi = W_i[idx];
        const float w = -(wi * wi);
        Wi2n[idx] = w;
        out_filters[SUB_NO * T_NO + idx] = w;       // rows 20..39
    } else if (g < KPAD_E + KPAD_I + SUB_NO * T_NO + SUB_NO) {
        const int s = g - (KPAD_E + KPAD_I + SUB_NO * T_NO);
        expW[s] = expf(W_sub[s + 1]);
    }
}

// ---------------- GEMM: syn[t, s] = sum_k S[t,k] * B[k, s]  via V_WMMA_F32_16X16X4_F32 ----
// Double-buffered LDS; B stored fragment-swizzled so each lane's (k,k+1) pair is one b64.
#define TM 128    // rows per block (8 waves x 16)
#define KC 64     // K chunk staged through LDS
#define APAD 4    // row pad: stride 68 floats -> conflict-free b64 fragment reads

__global__ __launch_bounds__(256) void syn_gemm_kernel(
    const float* __restrict__ S,     // [T, K]
    const float* __restrict__ Bm,    // [Kpad, NPAD], zero padded
    float* __restrict__ syn,         // [T, SUB_NO]
    int T, int K)
{
    __shared__ float As[2][TM][KC + APAD];      // 2 * 128*68*4 = 69632 B
    __shared__ float Bf[2][KC / 2][2 * NPAD];   // [k>>1][2n + (k&1)]; 2 * 32*64*4 = 16384 B

    const int tid  = threadIdx.x;
    const int lane = tid & 31;
    const int wave = tid >> 5;                 // 0..7
    const int blockRow = blockIdx.x * TM;
    const int waveRow  = wave * 16;

    v8f c0 = {}; v8f c1 = {};

    v4f aReg[8];
    v4f bReg[2];

    auto loadRegs = [&](int kb) {
        #pragma unroll
        for (int i = 0; i < 8; ++i) {
            const int idx = tid + i * 256;
            const int r  = idx >> 4;
            const int c4 = (idx & 15) << 2;
            const int gr = blockRow + r;
            const int gc = kb + c4;
            v4f v = {0.f, 0.f, 0.f, 0.f};
            if (gr < T && gc < K) {            // K is a multiple of 4 -> whole vec in-range
                // streamed exactly once -> non-temporal, keep L2 for reused buffers
                v = __builtin_nontemporal_load((const v4f*)(S + (size_t)gr * K + gc));
            }
            aReg[i] = v;
        }
        #pragma unroll
        for (int i = 0; i < 2; ++i) {
            const int idx = tid + i * 256;
            const int r  = idx >> 3;
            const int c4 = (idx & 7) << 2;
            bReg[i] = *(const v4f*)(Bm + (size_t)(kb + r) * NPAD + c4);
        }
    };

    auto storeLds = [&](int buf) {
        #pragma unroll
        for (int i = 0; i < 8; ++i) {
            const int idx = tid + i * 256;
            const int r  = idx >> 4;
            const int c4 = (idx & 15) << 2;
            *(v4f*)&As[buf][r][c4] = aReg[i];
        }
        #pragma unroll
        for (int i = 0; i < 2; ++i) {
            const int idx = tid + i * 256;
            const int k  = idx >> 3;
            const int n0 = (idx & 7) << 2;
            const float vv[4] = { bReg[i].x, bReg[i].y, bReg[i].z, bReg[i].w };
            #pragma unroll
            for (int j = 0; j < 4; ++j) {
                const int n = n0 + j;
                Bf[buf][k >> 1][2 * n + (k & 1)] = vv[j];
            }
        }
    };

    const int nChunks = (K + KC - 1) / KC;

    loadRegs(0);
    storeLds(0);
    __syncthreads();

    // A 16x4 f32 frag: lane<16 -> M=lane, K={k0,k0+1}; lane>=16 -> M=lane-16, K={k0+2,k0+3}
    // B 4x16 f32 frag mirrors with N on lanes.
    const int m     = lane & 15;
    const int khalf = (lane >> 4) << 1;        // 0 or 2

    for (int ch = 0; ch < nChunks; ++ch) {
        const int cur = ch & 1;
        const bool more = (ch + 1) < nChunks;
        if (more) loadRegs((ch + 1) * KC);      // global loads in flight during WMMA burst

        const float* Ab = &As[cur][waveRow + m][khalf];
        const float* Bb = &Bf[cur][khalf >> 1][2 * m];
        #pragma unroll
        for (int ks = 0; ks < KC; ks += 4) {
            const v2f a  = *(const v2f*)(Ab + ks);                    // ds b64
            const v2f b0 = *(const v2f*)(Bb + (ks >> 1) * 2 * NPAD);  // ds b64 (pairable
            const v2f b1 = *(const v2f*)(Bb + (ks >> 1) * 2 * NPAD + 2 * 16); // with b0)
            c0 = __builtin_amdgcn_wmma_f32_16x16x4_f32(false, a, false, b0, (short)0, c0, false, false);
            c1 = __builtin_amdgcn_wmma_f32_16x16x4_f32(false, a, false, b1, (short)0, c1, false, false);
        }

        if (more) {
            storeLds(cur ^ 1);
            __syncthreads();
        }
    }

    // D layout: c[r] at lane L<16 -> (M=r, N=L); L>=16 -> (M=r+8, N=L-16)
    const int n    = lane & 15;
    const int mAdd = (lane >> 4) * 8;
    #pragma unroll
    for (int r = 0; r < 8; ++r) {
        const int gm = blockRow + waveRow + r + mAdd;
        if (gm < T) {
            if (n < SUB_NO)      syn[(size_t)gm * SUB_NO + n]      = c0[r];
            if (16 + n < SUB_NO) syn[(size_t)gm * SUB_NO + 16 + n] = c1[r];
        }
    }
}

// ---------------- FIR (200-tap causal, depthwise) + tanh reduction ----------------
#define FT    256
#define HIST  (T_NO - 1)          // 199
#define FROWS (FT + HIST)         // 455
#define LSTR  21                  // gcd(21,64)==1 -> conflict-free LDS columns

__global__ __launch_bounds__(256) void fir_kernel(
    const float* __restrict__ synE, const float* __restrict__ synI,
    const float* __restrict__ We2,  const float* __restrict__ Wi2n,
    const float* __restrict__ Theta, const float* __restrict__ expW,
    const float* __restrict__ V_o, float* __restrict__ out, int T)
{
    __shared__ float sE[FROWS][LSTR];
    __shared__ float sI[FROWS][LSTR];
    __shared__ float kE[SUB_NO][T_NO];
    __shared__ float kI[SUB_NO][T_NO];

    const int tid = threadIdx.x;
    const int t0  = blockIdx.x * FT;
    const int r0  = t0 - HIST;

    for (int idx = tid; idx < FROWS * SUB_NO; idx += 256) {
        const int r = idx / SUB_NO;
        const int c = idx - r * SUB_NO;
        const int gt = r0 + r;
        float ve = 0.f, vi = 0.f;
        if (gt >= 0 && gt < T) {
            ve = synE[(size_t)gt * SUB_NO + c];
            vi = synI[(size_t)gt * SUB_NO + c];
        }
        sE[r][c] = ve;
        sI[r][c] = vi;
    }
    for (int idx = tid; idx < SUB_NO * T_NO; idx += 256) {
        const int s = idx / T_NO;
        const int k = idx - s * T_NO;
        kE[s][k] = We2[idx];
        kI[s][k] = Wi2n[idx];
    }
    __syncthreads();

    const int t    = t0 + tid;
    const int lrow = tid + HIST;
    float res = V_o[0];
    for (int s = 0; s < SUB_NO; ++s) {
        float acc = Theta[s + 1];
        #pragma unroll 8
        for (int k = 0; k < T_NO; ++k) {
            acc += sE[lrow - k][s] * kE[s][k];
            acc += sI[lrow - k][s] * kI[s][k];
        }
        res += tanhf(acc) * expW[s];
    }
    if (t < T) out[t] = res;
}

// ---------------- host-side orchestration ----------------
extern "C" void kernel_launch(void* const* d_in, const int* in_sizes, int n_in,
                              void* d_out, int out_size, void* d_ws, size_t ws_size,
                              hipStream_t stream) {
    (void)in_sizes; (void)n_in; (void)out_size; (void)ws_size;

    const float* S_e   = (const float*)d_in[0];
    const float* S_i   = (const float*)d_in[1];
    const float* W_e   = (const float*)d_in[2];
    const float* W_i   = (const float*)d_in[3];
    const float* Ce    = (const float*)d_in[4];
    const float* Ci    = (const float*)d_in[5];
    const float* W_sub = (const float*)d_in[6];
    const float* V_o   = (const float*)d_in[7];
    const float* Theta = (const float*)d_in[8];
    const float* temp  = (const float*)d_in[9];
    float* out = (float*)d_out;

    char* ws = (char*)d_ws;
    float* Bbuf_e = (float*)(ws + 0);            // 2048*32*4 = 262144
    float* Bbuf_i = (float*)(ws + 262144);       // 512*32*4  =  65536
    float* syn_e  = (float*)(ws + 327680);       // 20000*20*4 = 1600000
    float* syn_i  = (float*)(ws + 1927680);      // 1600000
    float* We2    = (float*)(ws + 3527680);      // 16000
    float* Wi2n   = (float*)(ws + 3543680);      // 16000
    float* expW   = (float*)(ws + 3559680);      // 80

    float* out_final   = out;                    // [20000]
    float* out_filters = out + T_DATA;           // [40*200]

    // 1) softmax mixing matrices (padded), squared FIR kernels, exp weights, out_filters
    {
        const int total = KPAD_E + KPAD_I + SUB_NO * T_NO + SUB_NO;  // 6580
        setup_kernel<<<(total + 255) / 256, 256, 0, stream>>>(
            W_e, W_i, Ce, Ci, W_sub, temp, Bbuf_e, Bbuf_i, We2, Wi2n, expW, out_filters);
    }

    // 2) WMMA GEMMs: syn_e = S_e @ C_e^T, syn_i = S_i @ C_i^T
    {
        const int grid = (T_DATA + TM - 1) / TM;  // 157
        syn_gemm_kernel<<<grid, 256, 0, stream>>>(S_e, Bbuf_e, syn_e, T_DATA, E_NO);
        syn_gemm_kernel<<<grid, 256, 0, stream>>>(S_i, Bbuf_i, syn_i, T_DATA, I_NO);
    }

    // 3) causal FIR + tanh reduction -> final
    {
        const int grid = (T_DATA + FT - 1) / FT;  // 79
        fir_kernel<<<grid, 256, 0, stream>>>(syn_e, syn_i, We2, Wi2n, Theta, expW, V_o,
                                             out_final, T_DATA);
    }
}